// CrossKTnet_55997783605736
// MI455X (gfx1250) — compile-verified
//
#include <hip/hip_runtime.h>
#include <math.h>

typedef float v2f __attribute__((ext_vector_type(2)));
typedef float v8f __attribute__((ext_vector_type(8)));

#define BN_SCALE_F 0.9999950000374998f   // 1/sqrt(1+1e-5)

// ---------------------------------------------------------------------------
// Generic batched WMMA GEMM (fp32, v_wmma_f32_16x16x4_f32)
//   D[z] = op( A[z/aDiv] * B[z] )
// AMODE template: 0 = A row-major contiguous K, 1 = A row-major strided K (aKs),
//                 2 = A transposed (A(m,k) = A[k*lda+m]).
// OOB rows/cols are address-clamped (never masked): garbage only lands in
// accumulator lanes the epilogue never stores. K must be a multiple of 4.
// Addressing: uniform scalar base pointer advances per k-step; per-lane offsets
// are loop-invariant 32-bit ints so loads can use the saddr+voffset (GVS) form.
// Block = 128 threads (4 waves), block tile 64(M) x 64(N), 32x32 per wave.
// ---------------------------------------------------------------------------
struct GemmParams {
  const float* A; const float* B; float* D; const float* bias;
  int M, N, K;
  int lda, ldb, ldd;
  long long sA, sB, sD;       // per-batch strides
  int aDiv;                   // A batch index = z / aDiv
  int aKs;                    // element stride along K in A rows (weight taps: 5)
  int cmV, cmStride, cmOff, cmTin;  // B column remap (conv); cmV==0 -> identity
  int flags;                  // 1 = accumulate into D, 2 = add bias[m]
  float scale;                // multiplied at the end (BN)
};

#if __has_builtin(__builtin_amdgcn_wmma_f32_16x16x4_f32)
#define WMMA_F32(a, b, c) __builtin_amdgcn_wmma_f32_16x16x4_f32(false, (a), false, (b), (short)0, (c), false, false)
#else
#define WMMA_F32(a, b, c) ((c) + (a).x * (b).x + (a).y * (b).y)  // placeholder fallback
#endif

template <int AMODE>
__global__ __launch_bounds__(128) void gemm_wmma_f32(GemmParams p) {
  const int lane  = threadIdx.x & 31;
  const int wave  = threadIdx.x >> 5;
  const int waveM = wave >> 1, waveN = wave & 1;
  const int half  = lane >> 4;
  const int r     = lane & 15;
  const int z     = blockIdx.z;
  const float* A  = p.A + (long long)(z / p.aDiv) * p.sA;
  const float* Bp = p.B + (long long)z * p.sB;
  float*       D  = p.D + (long long)z * p.sD;
  const int m0 = blockIdx.y * 64 + waveM * 32;
  const int n0 = blockIdx.x * 64 + waveN * 32;

  // per-lane rows / cols, clamped to safe in-bounds addresses
  int mrow[2], mA[2];
  int ncol[2], colB[2];
#pragma unroll
  for (int mt = 0; mt < 2; ++mt) {
    mrow[mt] = m0 + mt * 16 + r;
    mA[mt]   = mrow[mt] < p.M ? mrow[mt] : (p.M - 1);
  }
#pragma unroll
  for (int nt = 0; nt < 2; ++nt) {
    int n    = n0 + nt * 16 + r;
    ncol[nt] = n;
    int nc   = n < p.N ? n : (p.N - 1);
    int col;
    if (p.cmV > 0) {
      int t  = nc / p.cmV;
      int v  = nc - t * p.cmV;
      int ti = t * p.cmStride + p.cmOff;
      col = (ti >= 0 && ti < p.cmTin) ? (ti * p.cmV + v) : 0;
    } else {
      col = nc;
    }
    colB[nt] = col;
  }

  const int kb = half * 2;   // this lane's K offset within the 4-wide step

  // loop-invariant 32-bit lane offsets (elements); bases advance uniformly
  const float* aB = A;
  const float* bB = Bp;
  int aOff0, aOff1, aSec;
  long long aStep;
  if (AMODE == 2) {
    aOff0 = kb * p.lda + mA[0];
    aOff1 = kb * p.lda + mA[1];
    aSec  = p.lda;
    aStep = 4LL * p.lda;
  } else {
    aOff0 = mA[0] * p.lda + kb * p.aKs;
    aOff1 = mA[1] * p.lda + kb * p.aKs;
    aSec  = p.aKs;
    aStep = 4LL * p.aKs;
  }
  const int bOff0 = kb * p.ldb + colB[0];
  const int bOff1 = kb * p.ldb + colB[1];
  const long long bStep = 4LL * p.ldb;

  v8f acc[2][2] = {};
#pragma unroll 2
  for (int k0 = 0; k0 < p.K; k0 += 4) {
    v2f a0, a1, b0, b1;
    if (AMODE == 0) {           // contiguous pair -> single b64 load
      a0 = *(const v2f*)(aB + aOff0);
      a1 = *(const v2f*)(aB + aOff1);
    } else {                    // strided (taps) or transposed
      a0.x = aB[aOff0]; a0.y = aB[aOff0 + aSec];
      a1.x = aB[aOff1]; a1.y = aB[aOff1 + aSec];
    }
    b0.x = bB[bOff0]; b0.y = bB[bOff0 + p.ldb];
    b1.x = bB[bOff1]; b1.y = bB[bOff1 + p.ldb];
    acc[0][0] = WMMA_F32(a0, b0, acc[0][0]);
    acc[0][1] = WMMA_F32(a0, b1, acc[0][1]);
    acc[1][0] = WMMA_F32(a1, b0, acc[1][0]);
    acc[1][1] = WMMA_F32(a1, b1, acc[1][1]);
    aB += aStep; bB += bStep;
  }

#pragma unroll
  for (int mt = 0; mt < 2; ++mt) {
#pragma unroll
    for (int i = 0; i < 8; ++i) {
      int m = m0 + mt * 16 + half * 8 + i;
      if (m >= p.M) continue;
      float bi = (p.flags & 2) ? p.bias[m] : 0.f;
#pragma unroll
      for (int nt = 0; nt < 2; ++nt) {
        int n = ncol[nt];
        if (n >= p.N) continue;
        long long di = (long long)m * p.ldd + n;
        float val = acc[mt][nt][i];
        if (p.flags & 1) val += D[di];
        D[di] = (val + bi) * p.scale;
      }
    }
  }
}

// ---------------------------------------------------------------------------
// Elementwise / special kernels
// ---------------------------------------------------------------------------
__global__ void input_conv_relu(const float* __restrict__ x, const float* __restrict__ w,
                                const float* __restrict__ b, float* __restrict__ out,
                                int Nb, int T, int V) {
  long long idx = (long long)blockIdx.x * blockDim.x + threadIdx.x;
  long long total = (long long)Nb * 64 * T * V;
  if (idx >= total) return;
  int v = (int)(idx % V); long long q = idx / V;
  int t = (int)(q % T); q /= T;
  int co = (int)(q % 64); int n = (int)(q / 64);
  const float* xn = x + (long long)n * T * V;
  float s = b[co];
  for (int k = 0; k < 3; ++k) {
    int ti = t + k - 1;
    if (ti >= 0 && ti < T) s += w[co * 3 + k] * xn[(long long)ti * V + v];
  }
  s *= BN_SCALE_F;
  out[idx] = s > 0.f ? s : 0.f;
}

__global__ void pe_add(const float* __restrict__ x, float* __restrict__ y,
                       long long total, int C, int T, int V, float coef) {
  long long idx = (long long)blockIdx.x * blockDim.x + threadIdx.x;
  if (idx >= total) return;
  int v = (int)(idx % V); long long q = idx / V;
  q /= T;
  int c = (int)(q % C);
  int jj = c >> 1;
  float d = __expf((float)(2 * jj) * coef);
  float ang = (float)v * d;
  float pe = (c & 1) ? __cosf(ang) : __sinf(ang);
  y[idx] = x[idx] + pe;
}

__global__ void dism_from_gram(const float* __restrict__ g, float* __restrict__ dism,
                               int Nb, int V, float invCT2) {
  long long idx = (long long)blockIdx.x * blockDim.x + threadIdx.x;
  long long total = (long long)Nb * V * V;
  if (idx >= total) return;
  int u = (int)(idx % V); long long q = idx / V;
  int v = (int)(q % V); int n = (int)(q / V);
  const float* gn = g + (long long)n * V * V;
  float d2 = gn[(long long)v * V + v] + gn[(long long)u * V + u] - 2.f * gn[(long long)v * V + u];
  dism[idx] = fmaxf(d2, 0.f) * invCT2;
}

__global__ void reduce_max_stage1(const float* __restrict__ d, long long n, float* __restrict__ part) {
  __shared__ float s[256];
  float m = 0.f;
  for (long long i = (long long)blockIdx.x * 256 + threadIdx.x; i < n; i += (long long)gridDim.x * 256)
    m = fmaxf(m, d[i]);
  s[threadIdx.x] = m; __syncthreads();
  for (int off = 128; off > 0; off >>= 1) {
    if ((int)threadIdx.x < off) s[threadIdx.x] = fmaxf(s[threadIdx.x], s[threadIdx.x + off]);
    __syncthreads();
  }
  if (threadIdx.x == 0) part[blockIdx.x] = s[0];
}

__global__ void reduce_max_stage2(const float* __restrict__ part, int n, float* __restrict__ out) {
  __shared__ float s[256];
  float m = 0.f;
  for (int i = threadIdx.x; i < n; i += 256) m = fmaxf(m, part[i]);
  s[threadIdx.x] = m; __syncthreads();
  for (int off = 128; off > 0; off >>= 1) {
    if ((int)threadIdx.x < off) s[threadIdx.x] = fmaxf(s[threadIdx.x], s[threadIdx.x + off]);
    __syncthreads();
  }
  if (threadIdx.x == 0) out[0] = fmaxf(s[0], 1e-30f);
}

// sparsemax over the v-axis of exp(-exp(theta_s) * dism_norm); one block per (n,s,u)
__global__ __launch_bounds__(128) void attn_sparsemax(
    const float* __restrict__ dism, const float* __restrict__ maxval,
    const float* __restrict__ theta, const float* __restrict__ atts,
    const float* __restrict__ alphas, float* __restrict__ attn,
    int Nb, int S, int V) {
  int col = blockIdx.x;
  int u = col % V; int q = col / V;
  int s = q % S;   int n = q / S;
  int t = threadIdx.x;
  float invMax = 1.f / maxval[0];
  float dsexp = __expf(theta[s]);
  float z = 0.f;
  if (t < V) z = __expf(-dsexp * dism[((long long)n * V + t) * V + u] * invMax);

  __shared__ float zs[128];
  __shared__ float cs[128];
  __shared__ int   fl[128];
  zs[t] = z;
  __syncthreads();
  // bitonic sort descending (pads are 0 < any real exp() value -> sort last)
  for (int k = 2; k <= 128; k <<= 1) {
    for (int j = k >> 1; j > 0; j >>= 1) {
      int ixj = t ^ j;
      if (ixj > t) {
        float a = zs[t], c = zs[ixj];
        bool desc = ((t & k) == 0);
        if (desc ? (a < c) : (a > c)) { zs[t] = c; zs[ixj] = a; }
      }
      __syncthreads();
    }
  }
  // inclusive scan
  cs[t] = zs[t]; __syncthreads();
  for (int off = 1; off < 128; off <<= 1) {
    float add = (t >= off) ? cs[t - off] : 0.f;
    __syncthreads();
    cs[t] += add;
    __syncthreads();
  }
  int flag = 0;
  if (t < V) flag = (zs[t] * (float)(t + 1) > cs[t] - 1.f) ? 1 : 0;
  fl[t] = flag; __syncthreads();
  for (int off = 64; off > 0; off >>= 1) {
    if (t < off) fl[t] += fl[t + off];
    __syncthreads();
  }
  int ks = fl[0];
  float tau = (cs[ks - 1] - 1.f) / (float)ks;
  if (t < V) {
    float ka = fmaxf(z - tau, 0.f);
    long long o = (((long long)n * S + s) * V + t) * V + u;
    attn[o] = atts[((long long)s * V + t) * V + u] + ka * alphas[s];
  }
}

__global__ void add_lrelu(const float* __restrict__ a, const float* __restrict__ b,
                          float* __restrict__ out, long long n) {
  long long idx = (long long)blockIdx.x * blockDim.x + threadIdx.x;
  if (idx >= n) return;
  float s = a[idx] + b[idx];
  out[idx] = s >= 0.f ? s : 0.1f * s;
}

__global__ void pool_mean(const float* __restrict__ h, float* __restrict__ pooled, int TV) {
  __shared__ float s[256];
  const float* base = h + (long long)blockIdx.x * TV;
  float acc = 0.f;
  for (int i = threadIdx.x; i < TV; i += 256) acc += base[i];
  s[threadIdx.x] = acc; __syncthreads();
  for (int off = 128; off > 0; off >>= 1) {
    if ((int)threadIdx.x < off) s[threadIdx.x] += s[threadIdx.x + off];
    __syncthreads();
  }
  if (threadIdx.x == 0) pooled[blockIdx.x] = s[0] / (float)TV;
}

__global__ void fc_out(const float* __restrict__ pooled, const float* __restrict__ w,
                       const float* __restrict__ bias, float* __restrict__ out,
                       int Nb, int Cf, int O) {
  int i = blockIdx.x * blockDim.x + threadIdx.x;
  if (i >= Nb * O) return;
  int j = i % O, n = i / O;
  float s = bias[j];
  for (int c = 0; c < Cf; ++c) s += pooled[(long long)n * Cf + c] * w[(long long)j * Cf + c];
  out[i] = s;
}

// ---------------------------------------------------------------------------
// Host side
// ---------------------------------------------------------------------------
static void gemm(hipStream_t st, const float* A, const float* B, float* D, const float* bias,
                 int M, int N, int K, int lda, int ldb, int ldd,
                 long long sA, long long sB, long long sD, int batches,
                 int aDiv, int aKs, int transA,
                 int cmV, int cmStride, int cmOff, int cmTin,
                 int flags, float scale) {
  GemmParams p;
  p.A = A; p.B = B; p.D = D; p.bias = bias;
  p.M = M; p.N = N; p.K = K; p.lda = lda; p.ldb = ldb; p.ldd = ldd;
  p.sA = sA; p.sB = sB; p.sD = sD;
  p.aDiv = aDiv; p.aKs = aKs;
  p.cmV = cmV; p.cmStride = cmStride; p.cmOff = cmOff; p.cmTin = cmTin;
  p.flags = flags; p.scale = scale;
  dim3 g((unsigned)((N + 63) / 64), (unsigned)((M + 63) / 64), (unsigned)batches);
  if (transA)          gemm_wmma_f32<2><<<g, dim3(128), 0, st>>>(p);
  else if (aKs == 1)   gemm_wmma_f32<0><<<g, dim3(128), 0, st>>>(p);
  else                 gemm_wmma_f32<1><<<g, dim3(128), 0, st>>>(p);
}

struct Blk { const float *theta, *atts, *alphas, *vw, *vb, *ow, *ob, *fw, *fb, *d1w, *d1b, *d2w, *d2b; };

extern "C" void kernel_launch(void* const* d_in, const int* in_sizes, int n_in,
                              void* d_out, int out_size, void* d_ws, size_t ws_size,
                              hipStream_t stream) {
  const int NB = 32, V = 116, S = 3;
  if (n_in < 40) return;
  auto in = [&](int i) { return (const float*)d_in[i]; };

  const float *x_in, *input_w, *input_b, *fc_w, *fc_b;
  Blk blk[3];
  const bool downFlag[3] = {false, true, true};

  if (in_sizes[0] == 475136) {
    // insertion-order flattening: x, input_w, input_b, fc_w, fc_b, blocks...
    x_in = in(0); input_w = in(1); input_b = in(2); fc_w = in(3); fc_b = in(4);
    int i = 5;
    for (int b = 0; b < 3; ++b) {
      blk[b].theta = in(i++); blk[b].atts = in(i++); blk[b].alphas = in(i++);
      blk[b].vw = in(i++); blk[b].vb = in(i++);
      blk[b].ow = in(i++); blk[b].ob = in(i++);
      blk[b].fw = in(i++); blk[b].fb = in(i++);
      if (downFlag[b]) { blk[b].d1w = in(i++); blk[b].d1b = in(i++); blk[b].d2w = in(i++); blk[b].d2b = in(i++); }
      else { blk[b].d1w = blk[b].d1b = blk[b].d2w = blk[b].d2b = nullptr; }
    }
  } else {
    // alphabetical (tree-flatten) order: blocks(sorted keys), fc_b, fc_w, input_b, input_w, x
    int i = 0;
    for (int b = 0; b < 3; ++b) {
      blk[b].alphas = in(i++); blk[b].atts = in(i++);
      if (downFlag[b]) { blk[b].d1b = in(i++); blk[b].d1w = in(i++); blk[b].d2b = in(i++); blk[b].d2w = in(i++); }
      else { blk[b].d1w = blk[b].d1b = blk[b].d2w = blk[b].d2b = nullptr; }
      blk[b].fb = in(i++); blk[b].fw = in(i++);
      blk[b].ob = in(i++); blk[b].ow = in(i++);
      blk[b].theta = in(i++);
      blk[b].vb = in(i++); blk[b].vw = in(i++);
    }
    fc_b = in(i++); fc_w = in(i++); input_b = in(i++); input_w = in(i++); x_in = in(i++);
  }

  // workspace layout (floats)
  const long long SZ_ACT = (long long)NB * 8192 * V;   // 30,408,704 (same every block)
  const long long SZ_E   = (long long)S * 8192 * V;    // per-sample einsum buffer
  const long long SZ_VV  = (long long)NB * V * V;
  const long long SZ_ATT = (long long)NB * S * V * V;
  float* ws    = (float*)d_ws;
  float* bufX  = ws;
  float* bufY  = bufX + SZ_ACT;
  float* bufT1 = bufY + SZ_ACT;
  float* bufT2 = bufT1 + SZ_ACT;
  float* bufE  = bufT2 + SZ_ACT;
  float* bufG  = bufE + SZ_E;
  float* bufDm = bufG + SZ_VV;
  float* bufAt = bufDm + SZ_VV;
  float* bufP  = bufAt + SZ_ATT;         // 8192 pooled
  float* bufR  = bufP + 8192;            // 512 partials + 1 max
  long long needF = (bufR + 520) - ws;
  if ((long long)(ws_size / sizeof(float)) < needF) return;

  // ---- input conv: (32,1,128,116) -> (32,64,128,116), BN + ReLU
  {
    long long tot = SZ_ACT;
    input_conv_relu<<<(unsigned)((tot + 255) / 256), 256, 0, stream>>>(
        x_in, input_w, input_b, bufX, NB, 128, V);
  }

  const int cfg[3][3] = {{64, 64, 1}, {64, 128, 2}, {128, 256, 2}};
  int T = 128;
  for (int b = 0; b < 3; ++b) {
    const int cin = cfg[b][0], cout = cfg[b][1], stride = cfg[b][2];
    const int Tin = T, Tout = (Tin - 1) / stride + 1;
    const int CT = cin * Tin;               // = 8192 for every block
    const long long actIn  = (long long)NB * cin * Tin * V;
    const long long actOut = (long long)NB * cout * Tout * V;
    const Blk& p = blk[b];

    // 1. y_pe = x + PE  -> bufY
    pe_add<<<(unsigned)((actIn + 255) / 256), 256, 0, stream>>>(
        bufX, bufY, actIn, cin, Tin, V, -logf(10000.f) / (float)cin);

    // 2. Gram = Y^T Y per sample (WMMA, transA)
    gemm(stream, bufY, bufY, bufG, nullptr, V, V, CT, V, V, V,
         (long long)CT * V, (long long)CT * V, (long long)V * V, NB,
         1, 1, 1, 0, 0, 0, 0, 0, 1.f);

    // 3. dism
    dism_from_gram<<<(unsigned)((SZ_VV + 255) / 256), 256, 0, stream>>>(
        bufG, bufDm, NB, V, 1.f / ((float)CT * (float)CT));

    // 4. global max of dism
    reduce_max_stage1<<<512, 256, 0, stream>>>(bufDm, SZ_VV, bufR);
    reduce_max_stage2<<<1, 256, 0, stream>>>(bufR, 512, bufR + 512);

    // 5. attn = atts + sparsemax(exp(-exp(theta)*dism/max)) * alphas
    attn_sparsemax<<<(unsigned)(NB * S * V), 128, 0, stream>>>(
        bufDm, bufR + 512, p.theta, p.atts, p.alphas, bufAt, NB, S, V);

    // 6. vx = vmap 1x1 conv (x -> bufY, reuse)
    gemm(stream, p.vw, bufX, bufY, p.vb, cin, Tin * V, cin,
         cin, Tin * V, Tin * V, 0, (long long)cin * Tin * V, (long long)cin * Tin * V, NB,
         1, 1, 0, 0, 0, 0, 0, 2, 1.f);

    // 7. per-sample: einsum (WMMA) + 5-tap temporal conv (WMMA, implicit im2col)
    for (int n = 0; n < NB; ++n) {
      const float* vxn = bufY + (long long)n * CT * V;
      const float* atn = bufAt + (long long)n * S * V * V;
      // y[s, ct, v] = sum_u vx[ct,u] * attn[s,u,v]   (batched over s)
      gemm(stream, vxn, atn, bufE, nullptr, CT, V, V, V, V, V,
           0, (long long)V * V, (long long)CT * V, S,
           1, 1, 0, 0, 0, 0, 0, 0, 1.f);
      float* dOut = bufT1 + (long long)n * cout * Tout * V;
      for (int tap = 0; tap < 5; ++tap) {
        int flags = (tap ? 1 : 0) | (tap == 4 ? 2 : 0);
        float sc = (tap == 4) ? BN_SCALE_F : 1.f;
        gemm(stream, p.ow + tap, bufE, dOut, p.ob,
             cout, Tout * V, S * cin, S * cin * 5, Tin * V, Tout * V,
             0, 0, 0, 1,
             1, 5, 0, V, stride, tap - 2, Tin, flags, sc);
      }
    }

    // 8. residual 1 (down1 if needed) + leaky
    const float* res1 = bufX;
    if (downFlag[b]) {
      gemm(stream, p.d1w, bufX, bufT2, p.d1b, cout, Tout * V, cin,
           cin, Tin * V, Tout * V, 0, (long long)cin * Tin * V, (long long)cout * Tout * V, NB,
           1, 1, 0, V, stride, 0, Tin, 2, BN_SCALE_F);
      res1 = bufT2;
    }
    add_lrelu<<<(unsigned)((actOut + 255) / 256), 256, 0, stream>>>(res1, bufT1, bufT1, actOut);

    // 9. ff 1x1 conv + BN -> bufY
    gemm(stream, p.fw, bufT1, bufY, p.fb, cout, Tout * V, cout,
         cout, Tout * V, Tout * V, 0, (long long)cout * Tout * V, (long long)cout * Tout * V, NB,
         1, 1, 0, 0, 0, 0, 0, 2, BN_SCALE_F);

    // 10. residual 2 (down2 if needed) + leaky -> new x in bufX
    const float* res2 = bufX;
    if (downFlag[b]) {
      gemm(stream, p.d2w, bufX, bufT2, p.d2b, cout, Tout * V, cin,
           cin, Tin * V, Tout * V, 0, (long long)cin * Tin * V, (long long)cout * Tout * V, NB,
           1, 1, 0, V, stride, 0, Tin, 2, BN_SCALE_F);
      res2 = bufT2;
    }
    add_lrelu<<<(unsigned)((actOut + 255) / 256), 256, 0, stream>>>(res2, bufY, bufX, actOut);

    T = Tout;
  }

  // ---- global pool (mean over T*V) + FC
  pool_mean<<<(unsigned)(NB * 256), 256, 0, stream>>>(bufX, bufP, T * V);
  fc_out<<<1, 64, 0, stream>>>(bufP, fc_w, fc_b, (float*)d_out, NB, 256, 2);
}